// AdaptiveKPool2d_48369921687563
// MI455X (gfx1250) — compile-verified
//
#include <hip/hip_runtime.h>
#include <stdint.h>

// AdaptiveKPool2d: per (b,c) channel over 56x56=3136 f32 values:
//   m = max; count = #(x >= 0.1*m); k = clip(count,1,10); out = mean(top-k)
// Memory-bound: 102.8 MB @ 23.3 TB/s => ~4.4us floor. One wave32 per channel,
// async global->LDS DMA for staging, register-resident exact top-k.

#define HW        3136
#define PER_LANE  98          // 3136 / 32 lanes
#define WPB       4           // waves per block
#define CH_BYTES  (HW * 4)    // 12544 bytes per channel
#define MAX_K     10
#define THRESH    0.1f

#if defined(__has_builtin)
#if __has_builtin(__builtin_amdgcn_global_load_async_to_lds_b128)
#define HAVE_ASYNC 1
#endif
#endif
#ifndef HAVE_ASYNC
#define HAVE_ASYNC 0
#endif

typedef int v4i __attribute__((ext_vector_type(4)));           // 16-byte chunk
typedef __attribute__((address_space(1))) v4i* as1_v4i_p;      // global int4*
typedef __attribute__((address_space(3))) v4i* as3_v4i_p;      // LDS int4*
typedef __attribute__((address_space(3))) char* as3_char_p;
typedef __attribute__((address_space(1))) char* as1_char_p;

__device__ __forceinline__ float wave_max(float x) {
  #pragma unroll
  for (int m = 16; m > 0; m >>= 1) x = fmaxf(x, __shfl_xor(x, m, 32));
  return x;
}
__device__ __forceinline__ int wave_sum(int x) {
  #pragma unroll
  for (int m = 16; m > 0; m >>= 1) x += __shfl_xor(x, m, 32);
  return x;
}

__global__ __launch_bounds__(WPB * 32, 1)
void adaptive_kpool2d_kernel(const float* __restrict__ x, float* __restrict__ out) {
  __shared__ __align__(16) char smem[WPB * CH_BYTES];  // 50176 B / block

  const int lane = threadIdx.x & 31;
  const int warp = threadIdx.x >> 5;
  const int ch   = blockIdx.x * WPB + warp;            // 0..8191

  const float* gch = x + (size_t)ch * HW;
  char* sch = smem + warp * CH_BYTES;

  // ---- Stage channel into LDS -------------------------------------------
#if HAVE_ASYNC
  {
    // Global generic<->as(1) pointers are bit-identical; LDS needs a real
    // addrspacecast generic->as(3).
    as1_char_p gb = (as1_char_p)(uintptr_t)gch;
    as3_char_p lb = (as3_char_p)sch;
    // 784 float4 chunks: 24 full rounds of 32 lanes + 16-lane remainder.
    #pragma unroll
    for (int t = 0; t < 24; ++t) {
      const int off = (t * 32 + lane) * 16;
      __builtin_amdgcn_global_load_async_to_lds_b128(
          (as1_v4i_p)(gb + off), (as3_v4i_p)(lb + off), 0, 0);
    }
    if (lane < 16) {
      const int off = (24 * 32 + lane) * 16;
      __builtin_amdgcn_global_load_async_to_lds_b128(
          (as1_v4i_p)(gb + off), (as3_v4i_p)(lb + off), 0, 0);
    }
    asm volatile("s_wait_asynccnt 0" ::: "memory");
  }
#else
  {
    const float4* g4 = (const float4*)gch;
    float4* s4 = (float4*)sch;
    #pragma unroll
    for (int t = 0; t < 24; ++t) s4[t * 32 + lane] = g4[t * 32 + lane];
    if (lane < 16) s4[768 + lane] = g4[768 + lane];
  }
#endif

  // ---- LDS -> registers (lane-contiguous => bank-conflict-free) ---------
  float v[PER_LANE];
  {
    const float* s = (const float*)sch;
    #pragma unroll
    for (int j = 0; j < PER_LANE; ++j) v[j] = s[lane + 32 * j];
  }

  // ---- Pass 1: channel max ----------------------------------------------
  float m = -__builtin_inff();
  #pragma unroll
  for (int j = 0; j < PER_LANE; ++j) m = fmaxf(m, v[j]);
  m = wave_max(m);

  // ---- Pass 2: adaptive k = clip(count(x >= 0.1*max), 1, 10) ------------
  const float th = m * THRESH;
  int cnt = 0;
  #pragma unroll
  for (int j = 0; j < PER_LANE; ++j) cnt += (v[j] >= th) ? 1 : 0;
  cnt = wave_sum(cnt);
  const int k = (cnt < 1) ? 1 : ((cnt > MAX_K) ? MAX_K : cnt);

  // ---- Pass 3: exact sum of top-k (handles duplicate values) ------------
  float cur = m;
  float sum = 0.0f;
  int remaining = k;
  #pragma unroll 1
  for (int it = 0; it < MAX_K; ++it) {
    int meq = 0;
    #pragma unroll
    for (int j = 0; j < PER_LANE; ++j) meq += (v[j] == cur) ? 1 : 0;
    meq = wave_sum(meq);
    const int take = (meq < remaining) ? meq : remaining;
    sum += (float)take * cur;
    remaining -= take;
    if (remaining <= 0) break;
    float nm = -__builtin_inff();
    #pragma unroll
    for (int j = 0; j < PER_LANE; ++j) nm = (v[j] < cur) ? fmaxf(nm, v[j]) : nm;
    nm = wave_max(nm);
    cur = nm;
  }

  if (lane == 0) out[ch] = sum / (float)k;
}

extern "C" void kernel_launch(void* const* d_in, const int* in_sizes, int n_in,
                              void* d_out, int out_size, void* d_ws, size_t ws_size,
                              hipStream_t stream) {
  (void)n_in; (void)d_ws; (void)ws_size; (void)out_size;
  const float* x  = (const float*)d_in[0];
  float*      out = (float*)d_out;
  const int channels = in_sizes[0] / HW;           // 32*256 = 8192
  const int blocks   = (channels + WPB - 1) / WPB; // 2048
  adaptive_kpool2d_kernel<<<blocks, WPB * 32, 0, stream>>>(x, out);
}